// DSFusionv2_28819230556604
// MI455X (gfx1250) — compile-verified
//
#include <hip/hip_runtime.h>
#include <hip/hip_bf16.h>
#include <stdint.h>

// ---- static problem config (matches reference) ----
constexpr int kB = 2, kN = 6, kD = 48, kFH = 16, kFW = 44, kC = 80;
constexpr int kNX = 256, kNY = 256, kNZ = 1;
constexpr int kNPRIME = kB * kN * kD * kFH * kFW;          // 405504
constexpr size_t kACC_ELEMS = (size_t)kB * kNY * kNX * kC; // 10,485,760 floats
constexpr size_t kACC_BYTES = kACC_ELEMS * sizeof(float);  // ~41.9 MB (fits in 192MB L2)

typedef __attribute__((ext_vector_type(2))) float v2f;
typedef __attribute__((ext_vector_type(8))) float v8f;

// ---------------------------------------------------------------------------
// 3x3 inverse via adjugate
// ---------------------------------------------------------------------------
__device__ __forceinline__ void inv3(const float m[9], float o[9]) {
    float a = m[0], b = m[1], c = m[2];
    float d = m[3], e = m[4], f = m[5];
    float g = m[6], h = m[7], i = m[8];
    float A  =  (e * i - f * h);
    float Bq = -(d * i - f * g);
    float Cq =  (d * h - e * g);
    float det = a * A + b * Bq + c * Cq;
    float rd = 1.0f / det;
    o[0] = A * rd;              o[1] = -(b * i - c * h) * rd;  o[2] =  (b * f - c * e) * rd;
    o[3] = Bq * rd;             o[4] =  (a * i - c * g) * rd;  o[5] = -(a * f - c * d) * rd;
    o[6] = Cq * rd;             o[7] = -(a * h - b * g) * rd;  o[8] =  (a * e - b * d) * rd;
}

// ---------------------------------------------------------------------------
// Kernel 0: zero the channel-last accumulator (float4 stores, stays in L2)
// ---------------------------------------------------------------------------
__global__ void lss_zero_kernel(float4* __restrict__ acc4, int n4) {
    int i = blockIdx.x * blockDim.x + threadIdx.x;
    if (i < n4) acc4[i] = make_float4(0.f, 0.f, 0.f, 0.f);
}

// ---------------------------------------------------------------------------
// Kernel 1: geometry. One wave32 handles 16 frustum points; the ego-frame
// transform  E = combine @ Q + trans  is computed on the matrix engine via
// V_WMMA_F32_16X16X4_F32:
//   A (16x4)  rows 0..2 = [M row | t_r], rows 3..15 = 0
//             layout: VGPR0 = {A[r][0] (lanes 0-15), A[r][2] (lanes 16-31)}
//                     VGPR1 = {A[r][1]             , A[r][3]             }
//   B (4x16)  column p = (qx,qy,qz,1). Lanes 16-31 compute the SAME point as
//             lanes 0-15 (lane&15), so each lane supplies its own q values:
//             VGPR0 = {qx (K=0) | qz (K=2)}, VGPR1 = {qy (K=1) | 1.0 (K=3)}
//   D (16x16) rows 0..2 = (ex,ey,ez) for the 16 points -> lanes 0-15 read
//             d[0..2] directly and store the voxel index. No shuffles needed.
// EXEC is all-1s: the grid divides kNPRIME exactly, no guards before WMMA.
// ---------------------------------------------------------------------------
__global__ void lss_geom_wmma_kernel(const float* __restrict__ rots,
                                     const float* __restrict__ trans,
                                     const float* __restrict__ intrins,
                                     const float* __restrict__ post_rots,
                                     const float* __restrict__ post_trans,
                                     int* __restrict__ idxArr) {
    const int lane = threadIdx.x & 31;
    const int wid  = blockIdx.x * (blockDim.x >> 5) + (threadIdx.x >> 5);
    const int r    = lane & 15;          // point-in-group / A-matrix row
    const int pidx = wid * 16 + r;       // lanes 16-31 mirror lanes 0-15

    // decompose point index (16-point groups never cross a (b,n) slice:
    // D*FH*FW = 33792 is divisible by 16)
    int w = pidx % kFW;  int t = pidx / kFW;
    int h = t % kFH;     t /= kFH;
    int d = t % kD;      t /= kD;
    int n = t % kN;      int b = t / kN;
    int bn = b * kN + n;

    float PR[9], IPR[9], K[9], IK[9], R[9];
#pragma unroll
    for (int k = 0; k < 9; ++k) {
        PR[k] = post_rots[bn * 9 + k];
        K[k]  = intrins[bn * 9 + k];
        R[k]  = rots[bn * 9 + k];
    }
    inv3(PR, IPR);
    inv3(K, IK);
    // combine = rots @ inv(intrins)   (wave-uniform per (b,n))
    float M[9];
#pragma unroll
    for (int rr = 0; rr < 3; ++rr)
#pragma unroll
        for (int cc = 0; cc < 3; ++cc)
            M[rr * 3 + cc] = R[rr * 3 + 0] * IK[0 * 3 + cc] +
                             R[rr * 3 + 1] * IK[1 * 3 + cc] +
                             R[rr * 3 + 2] * IK[2 * 3 + cc];
    float tx = trans[bn * 3 + 0], ty = trans[bn * 3 + 1], tz = trans[bn * 3 + 2];

    // frustum: xs = linspace(0,703,44), ys = linspace(0,255,16), ds = 2 + i*56/48
    float fx = (float)w * (703.0f / 43.0f);
    float fy = (float)h * 17.0f;
    float fz = 2.0f + (float)d * (56.0f / 48.0f);

    float px = fx - post_trans[bn * 3 + 0];
    float py = fy - post_trans[bn * 3 + 1];
    float pz = fz - post_trans[bn * 3 + 2];

    float qx = IPR[0] * px + IPR[1] * py + IPR[2] * pz;
    float qy = IPR[3] * px + IPR[4] * py + IPR[5] * pz;
    float qz = IPR[6] * px + IPR[7] * py + IPR[8] * pz;
    qx *= qz;  qy *= qz;   // un-normalize pixel coords by depth

    // ---- assemble A (combine matrix + translation), per documented layout ----
    bool lo = (lane < 16);
    float a0 = (r < 3) ? (lo ? M[r * 3 + 0] : M[r * 3 + 2]) : 0.0f;
    float tr = (r == 0) ? tx : ((r == 1) ? ty : tz);
    float a1 = (r < 3) ? (lo ? M[r * 3 + 1] : tr) : 0.0f;
    v2f A;  A.x = a0;  A.y = a1;

    // ---- assemble B (point columns), no cross-lane movement needed ----
    v2f Bm;  Bm.x = lo ? qx : qz;  Bm.y = lo ? qy : 1.0f;

    v8f Cz = {};
    v8f Dv = __builtin_amdgcn_wmma_f32_16x16x4_f32(
        /*neg_a=*/false, A, /*neg_b=*/false, Bm,
        /*c_mod=*/(short)0, Cz, /*reuse_a=*/false, /*reuse_b=*/false);

    // rows 0..2, columns 0..15 live in VGPRs 0..2, lanes 0..15
    float ex = Dv[0], ey = Dv[1], ez = Dv[2];

    // voxelize: truncation toward zero matches astype(int32)
    int gx = (int)((ex - (-51.2f)) / 0.4f);
    int gy = (int)((ey - (-51.2f)) / 0.4f);
    int gz = (int)((ez - (-10.0f)) / 20.0f);

    bool kept = (gx >= 0) & (gx < kNX) & (gy >= 0) & (gy < kNY) &
                (gz >= 0) & (gz < kNZ);
    int flat = kept ? (((b * kNY + gy) * kNX + gx) * kC) : -1;

    if (lane < 16) idxArr[pidx] = flat;   // predication only after the WMMA
}

// ---------------------------------------------------------------------------
// Kernel 2: scatter — one wave (32 lanes) per point; lane covers channels
// c = lane, lane+32, lane+64. x is streamed once -> non-temporal loads so the
// 42MB accumulator stays resident in the 192MB L2 for the atomic RMWs.
// ---------------------------------------------------------------------------
__global__ void lss_scatter_kernel(const float* __restrict__ x,
                                   const int* __restrict__ idxArr,
                                   float* __restrict__ acc) {
    int wid  = blockIdx.x * (blockDim.x >> 5) + (threadIdx.x >> 5);
    int lane = threadIdx.x & 31;
    if (wid >= kNPRIME) return;

    int base = idxArr[wid];
    if (base < 0) return;

    const float* xp = x + (size_t)wid * kC;
#pragma unroll
    for (int c = lane; c < kC; c += 32) {
        float v = __builtin_nontemporal_load(&xp[c]);
        (void)__hip_atomic_fetch_add(&acc[(size_t)base + c], v,
                                     __ATOMIC_RELAXED, __HIP_MEMORY_SCOPE_AGENT);
    }
}

// ---------------------------------------------------------------------------
// Kernel 3: transpose acc[B][NY][NX][C] -> out[B][C][NZ=1][NY][NX].
// Per block: fixed (b,y), 32 consecutive x => 10240 contiguous bytes, staged
// into LDS with CDNA5 async copies (global_load_async_to_lds_b128, ASYNCcnt),
// then written out coalesced across x with non-temporal stores (write-once).
// ---------------------------------------------------------------------------
#define TP_THREADS 320
__global__ void lss_transpose_kernel(const float* __restrict__ acc,
                                     float* __restrict__ out) {
    __shared__ float tile[32 * 84];      // pitch 84: 16B-aligned, few conflicts
    const int x0 = blockIdx.x * 32;      // 8 tiles over NX
    const int y  = blockIdx.y;           // NY
    const int b  = blockIdx.z;           // B
    const int tid = threadIdx.x;

    unsigned rowbase = (unsigned)((((b * kNY + y) * kNX + x0) * kC) * 4);
    unsigned lds0 = (unsigned)(uintptr_t)(&tile[0]);
    unsigned long long sbase = (unsigned long long)(uintptr_t)acc;

    // 640 chunks of 16B: chunk = i*20 + j covers row i (x0+i), floats j*4..j*4+3
#pragma unroll
    for (int k = 0; k < 2; ++k) {
        int chunk = tid + k * TP_THREADS;
        int i = chunk / 20;
        int j = chunk % 20;
        unsigned goff = rowbase + (unsigned)((i * kC + j * 4) * 4);
        unsigned loff = lds0 + (unsigned)((i * 84 + j * 4) * 4);
        asm volatile("global_load_async_to_lds_b128 %0, %1, %2"
                     :: "v"(loff), "v"(goff), "s"(sbase)
                     : "memory");
    }
    asm volatile("s_wait_asynccnt 0" ::: "memory");
    __syncthreads();

    // out[((b*C + c)*NY + y)*NX + x0 + i] = tile[i][c]
    size_t obase = (size_t)b * kC * kNY * kNX + (size_t)y * kNX + x0;
#pragma unroll
    for (int idx = tid; idx < 32 * kC; idx += TP_THREADS) {
        int c = idx >> 5;
        int i = idx & 31;
        __builtin_nontemporal_store(tile[i * 84 + c],
                                    &out[obase + (size_t)c * (kNY * kNX) + i]);
    }
}

// ---------------------------------------------------------------------------
extern "C" void kernel_launch(void* const* d_in, const int* in_sizes, int n_in,
                              void* d_out, int out_size, void* d_ws, size_t ws_size,
                              hipStream_t stream) {
    const float* x          = (const float*)d_in[0];
    const float* rots       = (const float*)d_in[1];
    const float* trans      = (const float*)d_in[2];
    const float* intrins    = (const float*)d_in[3];
    const float* post_rots  = (const float*)d_in[4];
    const float* post_trans = (const float*)d_in[5];
    float* out = (float*)d_out;

    float* acc  = (float*)d_ws;                          // 41.9 MB channel-last accumulator
    int* idxArr = (int*)((char*)d_ws + kACC_BYTES);      // 1.6 MB per-point flat index

    // 0) zero accumulator (float4, fully divisible)
    {
        int n4 = (int)(kACC_ELEMS / 4);                  // 2,621,440
        lss_zero_kernel<<<n4 / 256, 256, 0, stream>>>((float4*)acc, n4);
    }
    // 1) geometry via WMMA: 16 points per wave, 8 waves per block
    //    waves = 405504/16 = 25344 -> 3168 blocks (exact, EXEC always full)
    lss_geom_wmma_kernel<<<(kNPRIME / 16) / 8, 256, 0, stream>>>(
        rots, trans, intrins, post_rots, post_trans, idxArr);
    // 2) atomic scatter, one wave32 per point (8 waves / 256-thread block)
    lss_scatter_kernel<<<kNPRIME / 8, 256, 0, stream>>>(x, idxArr, acc);
    // 3) async-LDS tiled transpose to final layout
    lss_transpose_kernel<<<dim3(kNX / 32, kNY, kB), TP_THREADS, 0, stream>>>(acc, out);
}